// Fore_Back_dis_16355235463802
// MI455X (gfx1250) — compile-verified
//
#include <hip/hip_runtime.h>
#include <hip/hip_bf16.h>
#include <stdint.h>

// Problem constants (from reference): hsi[B=4, C=128, H=256, W=256], k = HW/2.
#define NB 4
#define NC 128
#define HW 65536
#define KSEL 32768
#define ROWS (NB * NC)      // 512 (c,b) rows, row id = b*NC + c (memory order)
#define BLK 256
#define ITERS (HW / (4 * BLK)) // 64 float4 iterations per thread

typedef float v2f __attribute__((ext_vector_type(2)));
typedef float v8f __attribute__((ext_vector_type(8)));
typedef unsigned int uint32;

__device__ __forceinline__ float entropy_of(float x, float w) {
    float f1 = x * w;
    return -f1 * __logf(f1); // f1 in (1e-5, 1) -> e > 0, so uint bit order == value order
}

// Wave-level 32-lane sum on the CDNA5 WMMA unit.
// A is the 16x4 f32 matrix: lanes 0-15 VGPR0 -> K=0 (rows 0..15),
// lanes 16-31 VGPR0 -> K=2; VGPR1 (K=1/K=3) zeroed. B = ones.
// D[m][*] = partial[m] + partial[m+16]; D vgpr r holds row r (lanes<16) / row 8+r.
// In-lane v8 sum gives sum(rows 0..7) on lanes<16 and sum(rows 8..15) on lanes>=16;
// one xor-16 lane exchange completes the wave sum.
__device__ __forceinline__ float wave_sum_wmma(float v) {
    v2f a; a.x = v;    a.y = 0.0f;
    v2f b; b.x = 1.0f; b.y = 1.0f;
    v8f c = {0.f, 0.f, 0.f, 0.f, 0.f, 0.f, 0.f, 0.f};
    c = __builtin_amdgcn_wmma_f32_16x16x4_f32(false, a, false, b, (short)0, c,
                                              false, false);
    float s = ((c[0] + c[1]) + (c[2] + c[3])) + ((c[4] + c[5]) + (c[6] + c[7]));
    s += __shfl_xor(s, 16, 32);
    return s;
}

// Block-wide sum; valid on thread 0 only.
__device__ __forceinline__ float block_sum(float v, float* wscratch) {
    float s = wave_sum_wmma(v);
    int lane = (int)threadIdx.x & 31;
    int wid  = (int)threadIdx.x >> 5;
    if (lane == 0) wscratch[wid] = s;
    __syncthreads();
    float r = 0.f;
    if (threadIdx.x == 0) {
        int nw = (int)blockDim.x >> 5;
        for (int i = 0; i < nw; ++i) r += wscratch[i];
    }
    return r;
}

// Descending radix scan over an LDS histogram: find bin B and residual rank r
// such that (count of bins > B) < target <= (count of bins >= B),
// r = target - count(bins > B).  Results broadcast via bcast[0]=B, bcast[1]=r.
__device__ __forceinline__ void scan_desc(const uint32* hist, int bins,
                                          uint32 target, uint32* locals,
                                          uint32* bcast) {
    const int T = (int)blockDim.x;
    const int t = (int)threadIdx.x;
    const int bpt = bins / T;
    const int lo = t * bpt;
    uint32 local = 0;
    for (int i = 0; i < bpt; ++i) local += hist[lo + i];
    locals[t] = local;
    __syncthreads();
    uint32 suff = 0;
    for (int tt = t + 1; tt < T; ++tt) suff += locals[tt]; // exclusive suffix-from-top
    if (suff < target && suff + local >= target) {         // exactly one thread
        uint32 acc = suff;
        for (int bin = lo + bpt - 1; bin >= lo; --bin) {
            uint32 cb = hist[bin];
            if (acc + cb >= target) { bcast[0] = (uint32)bin; bcast[1] = target - acc; break; }
            acc += cb;
        }
    }
    __syncthreads();
}

// ---- Pass 1: entropy + 12-bit (bits 31:20) count histogram + row total ----
__global__ __launch_bounds__(BLK) void k_pass1(const float* __restrict__ hsi,
                                               const float* __restrict__ wp,
                                               float* __restrict__ total,
                                               uint32* __restrict__ b1,
                                               uint32* __restrict__ r1) {
    __shared__ uint32 hist[4096];
    __shared__ uint32 locals[BLK];
    __shared__ uint32 bcast[2];
    __shared__ float  wscratch[BLK / 32];
    const int row = (int)blockIdx.x;
    const float w = wp[0];
    const float4* p = (const float4*)(hsi + (size_t)row * HW);
    for (int i = threadIdx.x; i < 4096; i += BLK) hist[i] = 0u;
    __syncthreads();

    float acc = 0.f;
    for (int it = 0; it < ITERS; ++it) {
        int idx = it * BLK + (int)threadIdx.x;
        if (it + 8 < ITERS) __builtin_prefetch(&p[idx + 8 * BLK], 0, 0);
        float4 x = p[idx];
        float e0 = entropy_of(x.x, w), e1 = entropy_of(x.y, w);
        float e2 = entropy_of(x.z, w), e3 = entropy_of(x.w, w);
        acc += (e0 + e1) + (e2 + e3);
        atomicAdd(&hist[__float_as_uint(e0) >> 20], 1u);
        atomicAdd(&hist[__float_as_uint(e1) >> 20], 1u);
        atomicAdd(&hist[__float_as_uint(e2) >> 20], 1u);
        atomicAdd(&hist[__float_as_uint(e3) >> 20], 1u);
    }
    __syncthreads();

    scan_desc(hist, 4096, (uint32)KSEL, locals, bcast);
    float ts = block_sum(acc, wscratch);
    if (threadIdx.x == 0) {
        total[row] = ts;
        b1[row] = bcast[0];
        r1[row] = bcast[1];
    }
}

// ---- Pass 2 (final data pass): within one streaming read, compute
//   accgt = sum(e | bin1 > B1)                       (registers)
//   count + f32-sum histograms over bits 19:8        (only bin1 == B1 elems)
// then scan for B2/r2, take S2 = sum of bin-sums strictly above B2, and
// approximate the r2 tie elements inside the 24-bit-prefix bin by the bin
// midpoint (relative error <= 2^-15, below f32 accumulation noise).
//   sum_high = accgt + S2 + r2 * mid(B1,B2) ----
__global__ __launch_bounds__(BLK) void k_pass2(const float* __restrict__ hsi,
                                               const float* __restrict__ wp,
                                               const uint32* __restrict__ b1p,
                                               const uint32* __restrict__ r1p,
                                               float* __restrict__ SH) {
    __shared__ uint32 hist[4096];
    __shared__ float  hsum[4096];
    __shared__ uint32 locals[BLK];
    __shared__ uint32 bcast[2];
    __shared__ float  wscratch[BLK / 32];
    const int row = (int)blockIdx.x;
    const float w = wp[0];
    const uint32 B1 = b1p[row];
    const uint32 target = r1p[row];
    const float4* p = (const float4*)(hsi + (size_t)row * HW);
    for (int i = threadIdx.x; i < 4096; i += BLK) { hist[i] = 0u; hsum[i] = 0.f; }
    __syncthreads();

    float accgt = 0.f;
    for (int it = 0; it < ITERS; ++it) {
        int idx = it * BLK + (int)threadIdx.x;
        if (it + 8 < ITERS) __builtin_prefetch(&p[idx + 8 * BLK], 0, 0);
        float4 x = p[idx];
        const float ev[4] = {entropy_of(x.x, w), entropy_of(x.y, w),
                             entropy_of(x.z, w), entropy_of(x.w, w)};
#pragma unroll
        for (int j = 0; j < 4; ++j) {
            uint32 u = __float_as_uint(ev[j]);
            uint32 bb = u >> 20;
            if (bb > B1) {
                accgt += ev[j];
            } else if (bb == B1) {
                uint32 bin = (u >> 8) & 0xFFFu;
                atomicAdd(&hist[bin], 1u);
                atomicAdd(&hsum[bin], ev[j]);
            }
        }
    }
    __syncthreads();

    scan_desc(hist, 4096, target, locals, bcast);
    const uint32 B2 = bcast[0];
    const uint32 rr2 = bcast[1];

    // Per-thread partial of S2 = sum over bins strictly above B2.
    float s2part = 0.f;
    {
        const int bpt = 4096 / BLK;
        const int lo = (int)threadIdx.x * bpt;
#pragma unroll
        for (int i = 0; i < bpt; ++i) {
            int bin = lo + i;
            if ((uint32)bin > B2) s2part += hsum[bin];
        }
    }
    float s = block_sum(accgt + s2part, wscratch);
    if (threadIdx.x == 0) {
        // Midpoint of the 24-bit-prefix bin holding the k-th value.
        float v = __uint_as_float((B1 << 20) | (B2 << 8) | 0x80u);
        SH[row] = s + (float)rr2 * v;
    }
}

// ---- Final: delta_c proportional to 2*sum_high - total, stable top-3 ----
__global__ void k_final(const float* __restrict__ total,
                        const float* __restrict__ SH,
                        int* __restrict__ out) {
    __shared__ float delta[NC];
    int c = (int)threadIdx.x;
    if (c < NC) {
        float sh = 0.f, tt = 0.f;
        for (int b = 0; b < NB; ++b) {
            int row = b * NC + c;
            sh += SH[row];
            tt += total[row];
        }
        delta[c] = 2.f * sh - tt;
    }
    __syncthreads();
    if (c == 0) {
        float v0 = -3.4e38f, v1 = -3.4e38f, v2 = -3.4e38f;
        int i0 = 0, i1 = 0, i2 = 0;
        for (int i = 0; i < NC; ++i) {
            float d = delta[i];
            if (d > v0)      { v2 = v1; i2 = i1; v1 = v0; i1 = i0; v0 = d; i0 = i; }
            else if (d > v1) { v2 = v1; i2 = i1; v1 = d;  i1 = i; }
            else if (d > v2) { v2 = d;  i2 = i; }
        }
        out[0] = i0; out[1] = i1; out[2] = i2;
    }
}

extern "C" void kernel_launch(void* const* d_in, const int* in_sizes, int n_in,
                              void* d_out, int out_size, void* d_ws, size_t ws_size,
                              hipStream_t stream) {
    (void)in_sizes; (void)n_in; (void)out_size; (void)ws_size;
    const float* hsi = (const float*)d_in[0];
    const float* w   = (const float*)d_in[1];

    // ~8 KB of per-row state in scratch.
    float*  total = (float*)d_ws;
    float*  SH    = total + ROWS;
    uint32* b1    = (uint32*)(SH + ROWS);
    uint32* r1    = b1 + ROWS;

    k_pass1<<<ROWS, BLK, 0, stream>>>(hsi, w, total, b1, r1);
    k_pass2<<<ROWS, BLK, 0, stream>>>(hsi, w, b1, r1, SH);
    k_final<<<1, NC, 0, stream>>>(total, SH, (int*)d_out);
}